// EarthMoverLoss_46377056862525
// MI455X (gfx1250) — compile-verified
//
#include <hip/hip_runtime.h>
#include <math.h>

typedef float v2f __attribute__((ext_vector_type(2)));
typedef float v8f __attribute__((ext_vector_type(8)));

#define NPTS 256
#define NBATCH 8
#define BIG 1e30f

// Wave-local LDS fence: one wave talking to itself through LDS.
// CDNA5 split counters: DS ops tracked by DScnt; s_wait_dscnt 0 guarantees
// all prior LDS writes from this wave are visible to subsequent LDS reads.
__device__ __forceinline__ void lds_fence_wave() {
    __builtin_amdgcn_wave_barrier();
    asm volatile("s_wait_dscnt 0" ::: "memory");
    __builtin_amdgcn_wave_barrier();
}

// Raw v_sqrt_f32 (~1 ULP): precision-neutral here (reference runs in f64;
// our f32 cost pipeline dominates the error budget), avoids the ~14-VALU
// IEEE-correct refinement sequence per element.
__device__ __forceinline__ float fast_sqrtf(float x) {
    return __builtin_amdgcn_sqrtf(x);
}

extern "C" __global__ void emd_zero_out(float* out) {
    if (threadIdx.x == 0 && blockIdx.x == 0) out[0] = 0.0f;
}

extern "C" __global__ __launch_bounds__(256)
void emd_hungarian_kernel(const float* __restrict__ pred,
                          const float* __restrict__ label,
                          float* __restrict__ out)
{
    extern __shared__ float smem[];
    float* cost      = smem;                    // NPTS*NPTS  (256 KB)
    float* xsq       = cost + NPTS * NPTS;      // NPTS
    float* ysq       = xsq + NPTS;              // NPTS
    float* u         = ysq + NPTS;              // NPTS  (row duals)
    float* vdual     = u + NPTS;                // NPTS  (col duals)
    float* shortest  = vdual + NPTS;            // NPTS
    int*   path      = (int*)(shortest + NPTS); // NPTS
    int*   col4row   = path + NPTS;             // NPTS
    int*   row4col   = col4row + NPTS;          // NPTS
    int*   sc        = row4col + NPTS;          // NPTS  (scanned cols)
    int*   sr        = sc + NPTS;               // NPTS  (scanned rows)
    float* redf      = (float*)(sr + NPTS);     // 8     (block reduce)

    const int b    = blockIdx.x;
    const int tid  = threadIdx.x;
    const int lane = tid & 31;
    const int wave = tid >> 5;

    const float* X = pred  + (size_t)b * NPTS * 3;
    const float* Y = label + (size_t)b * NPTS * 3;

    // ---- Phase 0: squared norms + state init (256 threads) ----
    {
        float x0 = X[tid * 3 + 0], x1 = X[tid * 3 + 1], x2 = X[tid * 3 + 2];
        xsq[tid] = x0 * x0 + x1 * x1 + x2 * x2;
        float y0 = Y[tid * 3 + 0], y1 = Y[tid * 3 + 1], y2 = Y[tid * 3 + 2];
        ysq[tid] = y0 * y0 + y1 * y1 + y2 * y2;
        u[tid] = 0.0f;
        vdual[tid] = 0.0f;
        col4row[tid] = -1;
        row4col[tid] = -1;
    }
    __syncthreads();

    // ---- Phase 1: cost matrix via V_WMMA_F32_16X16X4_F32 (all 8 waves) ----
    // cost[m][n] = sqrt(|x_m|^2 + |y_n|^2 - 2 * dot(x_m, y_n)); dot via WMMA,
    // K=3 padded to 4 with zeros. Wave w owns row-tiles tm = 2w, 2w+1 and
    // sweeps all 16 col-tiles; A fragment + xsq hoisted out of the inner loop.
    {
        const int hi  = lane >> 4;   // 0: K slots 0,1  | 1: K slots 2,3
        const int l15 = lane & 15;
        for (int tmi = 0; tmi < 2; ++tmi) {
            const int tm = wave * 2 + tmi;
            const int ar = tm * 16 + l15;        // A row (pred index)
            v2f a;
            a.x = hi ? X[ar * 3 + 2] : X[ar * 3 + 0];
            a.y = hi ? 0.0f          : X[ar * 3 + 1];
            float xs[8];
#pragma unroll
            for (int vv = 0; vv < 8; ++vv)
                xs[vv] = xsq[tm * 16 + vv + hi * 8];

            for (int tn = 0; tn < 16; ++tn) {
                const int bc = tn * 16 + l15;    // B col (label index)
                v2f bf;
                bf.x = hi ? Y[bc * 3 + 2] : Y[bc * 3 + 0];
                bf.y = hi ? 0.0f          : Y[bc * 3 + 1];
                const float ys = ysq[bc];        // this lane's column norm
                v8f c = {};
                c = __builtin_amdgcn_wmma_f32_16x16x4_f32(
                        /*neg_a=*/false, a, /*neg_b=*/false, bf,
                        /*c_mod=*/(short)0, c, /*reuse_a=*/false, /*reuse_b=*/false);
                float* crow = cost + (tm * 16 + hi * 8) * NPTS + bc;
#pragma unroll
                for (int vv = 0; vv < 8; ++vv) {
                    float d2 = xs[vv] + ys - 2.0f * c[vv];
                    crow[vv * NPTS] = fast_sqrtf(fmaxf(d2, 0.0f));
                }
            }
        }
    }
    __syncthreads();

    // ---- Phase 2: Hungarian (shortest augmenting path), wave 0 only ----
    // Lane `lane` owns columns j = k*32 + lane, k = 0..7.
    if (wave == 0) {
        for (int cur = 0; cur < NPTS; ++cur) {
#pragma unroll
            for (int k = 0; k < 8; ++k) {
                int j = k * 32 + lane;
                shortest[j] = BIG;
                path[j] = -1;
                sc[j] = 0;
                sr[j] = 0;
            }
            lds_fence_wave();

            float minv = 0.0f;
            int i = cur;
            int sink = -1;
            while (sink == -1) {
                if (lane == 0) sr[i] = 1;
                float ui = u[i];
                const float* crow = cost + i * NPTS;
#pragma unroll
                for (int k = 0; k < 8; ++k) {
                    int j = k * 32 + lane;
                    if (!sc[j]) {
                        float r = minv + crow[j] - ui - vdual[j];
                        if (r < shortest[j]) { shortest[j] = r; path[j] = i; }
                    }
                }
                lds_fence_wave();

                // argmin over unscanned columns (lane-local then wave reduce)
                float bv = BIG;
                int   bj = 0x7fffffff;
#pragma unroll
                for (int k = 0; k < 8; ++k) {
                    int j = k * 32 + lane;
                    float s = sc[j] ? BIG : shortest[j];
                    if (s < bv) { bv = s; bj = j; }
                }
                for (int off = 16; off > 0; off >>= 1) {
                    float ov = __shfl_xor(bv, off, 32);
                    int   oj = __shfl_xor(bj, off, 32);
                    if (ov < bv || (ov == bv && oj < bj)) { bv = ov; bj = oj; }
                }
                minv = bv;          // uniform across wave
                int j = bj;         // uniform across wave
                if (lane == 0) sc[j] = 1;
                int r4c = row4col[j];
                if (r4c == -1) sink = j;
                else           i = r4c;
                lds_fence_wave();
            }

            // dual updates (pre-augment col4row, as in reference)
#pragma unroll
            for (int k = 0; k < 8; ++k) {
                int r = k * 32 + lane;
                if (sr[r] && r != cur) u[r] += minv - shortest[col4row[r]];
            }
            if (lane == 0) u[cur] += minv;
#pragma unroll
            for (int k = 0; k < 8; ++k) {
                int j = k * 32 + lane;
                if (sc[j]) vdual[j] -= minv - shortest[j];
            }
            lds_fence_wave();

            // augment along alternating path (serial, lane 0)
            if (lane == 0) {
                int j = sink;
                while (true) {
                    int pi = path[j];
                    row4col[j] = pi;
                    int nj = col4row[pi];
                    col4row[pi] = j;
                    j = nj;
                    if (pi == cur) break;
                }
            }
            lds_fence_wave();
        }
    }
    __syncthreads();

    // ---- Phase 3: loss = mean_i || x_i - y_{col4row[i]} || ----
    {
        int m = col4row[tid];
        float dx = X[tid * 3 + 0] - Y[m * 3 + 0];
        float dy = X[tid * 3 + 1] - Y[m * 3 + 1];
        float dz = X[tid * 3 + 2] - Y[m * 3 + 2];
        float nrm = fast_sqrtf(dx * dx + dy * dy + dz * dz);
        for (int off = 16; off > 0; off >>= 1)
            nrm += __shfl_xor(nrm, off, 32);
        if (lane == 0) redf[wave] = nrm;
        __syncthreads();
        if (tid == 0) {
            float s = 0.0f;
#pragma unroll
            for (int w = 0; w < 8; ++w) s += redf[w];
            atomicAdd(out, s * (1.0f / (float)(NBATCH * NPTS)));
        }
    }
}

extern "C" void kernel_launch(void* const* d_in, const int* in_sizes, int n_in,
                              void* d_out, int out_size, void* d_ws, size_t ws_size,
                              hipStream_t stream) {
    (void)in_sizes; (void)n_in; (void)out_size; (void)d_ws; (void)ws_size;
    const float* pred  = (const float*)d_in[0];
    const float* label = (const float*)d_in[1];
    float* out = (float*)d_out;

    // LDS: cost (256KB) + 5 float[256] + 5 int[256] + 8 floats
    size_t shmem = (size_t)NPTS * NPTS * 4 + 10 * NPTS * 4 + 8 * 4;
    (void)hipFuncSetAttribute((const void*)emd_hungarian_kernel,
                              hipFuncAttributeMaxDynamicSharedMemorySize,
                              (int)shmem);

    emd_zero_out<<<1, 1, 0, stream>>>(out);
    emd_hungarian_kernel<<<NBATCH, 256, shmem, stream>>>(pred, label, out);
}